// FPGCN_90254442758730
// MI455X (gfx1250) — compile-verified
//
#include <hip/hip_runtime.h>
#include <hip/hip_bf16.h>

typedef __attribute__((ext_vector_type(2))) float v2f;
typedef __attribute__((ext_vector_type(8))) float v8f;

#define NNODES 50000
#define NEDGES 800000
#define DIM    128
#define DOUTF  64

// ---------------- elementwise / precompute kernels ----------------

__global__ void zero_f32(float* __restrict__ p, int n) {
    int i = blockIdx.x * blockDim.x + threadIdx.x;
    if (i < n) p[i] = 0.0f;
}

__global__ void deg_kernel(const int* __restrict__ col, float* __restrict__ deg, int e) {
    int i = blockIdx.x * blockDim.x + threadIdx.x;
    if (i < e)
        __hip_atomic_fetch_add(&deg[col[i]], 1.0f, __ATOMIC_RELAXED, __HIP_MEMORY_SCOPE_AGENT);
}

__global__ void dinv_kernel(float* __restrict__ deg, int n) {
    int i = blockIdx.x * blockDim.x + threadIdx.x;
    if (i < n) {
        float d = deg[i];
        deg[i] = (d > 0.0f) ? rsqrtf(d) : 0.0f;   // in-place deg -> d^{-1/2}
    }
}

__global__ void norm_kernel(const int* __restrict__ row, const int* __restrict__ col,
                            const float* __restrict__ dinv, float* __restrict__ norm, int e) {
    int i = blockIdx.x * blockDim.x + threadIdx.x;
    if (i < e) norm[i] = dinv[row[i]] * dinv[col[i]];
}

// x_tilde = M ? x : (use_h ? h : 0)
__global__ void xtilde_kernel(const unsigned char* __restrict__ M, const float* __restrict__ x,
                              const float* __restrict__ h, float* __restrict__ xt,
                              int n, int use_h) {
    int i = blockIdx.x * blockDim.x + threadIdx.x;
    if (i < n) xt[i] = M[i] ? x[i] : (use_h ? h[i] : 0.0f);
}

// one wave32 per edge: lane L handles channels [4L, 4L+4)  (32*4 = 128 = DIM)
__global__ void scatter_kernel(const int* __restrict__ row, const int* __restrict__ col,
                               const float* __restrict__ norm, const float* __restrict__ xt,
                               float* __restrict__ agg, int e) {
    int wid  = (blockIdx.x * blockDim.x + threadIdx.x) >> 5;
    int lane = threadIdx.x & 31;
    if (wid >= e) return;
    int   r = row[wid];
    int   c = col[wid];
    float w = norm[wid];
    const float4* src = (const float4*)(xt + (size_t)r * DIM);
    float4 v = src[lane];                       // 16B vector gather
    float* dst = agg + (size_t)c * DIM + lane * 4;
    __hip_atomic_fetch_add(dst + 0, w * v.x, __ATOMIC_RELAXED, __HIP_MEMORY_SCOPE_AGENT);
    __hip_atomic_fetch_add(dst + 1, w * v.y, __ATOMIC_RELAXED, __HIP_MEMORY_SCOPE_AGENT);
    __hip_atomic_fetch_add(dst + 2, w * v.z, __ATOMIC_RELAXED, __HIP_MEMORY_SCOPE_AGENT);
    __hip_atomic_fetch_add(dst + 3, w * v.w, __ATOMIC_RELAXED, __HIP_MEMORY_SCOPE_AGENT);
}

// ---------------- fp32 WMMA GEMM: out = A @ W^T + b1 (+ b2), optional ReLU ----
// A:[M,K] row-major, W:[dout,K] row-major (torch Linear), out:[M,dout].
// One wave computes a 16x16 tile via 32x V_WMMA_F32_16X16X4_F32.
//
// A 16x4 f32 layout (ISA 7.12.2): lane L holds rows m=L%16; VGPR v holds
// K = 2*(L/16) + v. B 4x16 mirrors this with N = L%16. C/D: VGPR v holds
// (row = v + 8*(L/16), col = L%16).
__global__ void gemm_wmma_f32(const float* __restrict__ A, const float* __restrict__ W,
                              const float* __restrict__ b1, const float* __restrict__ b2,
                              float* __restrict__ out,
                              int tiles_m, int tiles_o, int K, int dout, int do_relu) {
    int wave = (blockIdx.x * blockDim.x + threadIdx.x) >> 5;   // uniform per wave
    int lane = threadIdx.x & 31;
    int tm = wave / tiles_o;
    int to = wave - tm * tiles_o;
    if (tm >= tiles_m) return;                                  // wave-uniform exit

    int m    = lane & 15;      // A row within tile == B/C column within tile
    int half = lane >> 4;      // 0: K=0,1  1: K=2,3
    int kb   = half << 1;

    const float* arow = A + (size_t)(tm * 16 + m) * K + kb;     // node row
    const float* wrow = W + (size_t)(to * 16 + m) * K + kb;     // weight row (o = to*16+m)

    int   ocol = to * 16 + m;
    float bsum = b1[ocol] + (b2 ? b2[ocol] : 0.0f);
    v8f c = {bsum, bsum, bsum, bsum, bsum, bsum, bsum, bsum};

#pragma unroll 8
    for (int k = 0; k < K; k += 4) {
        v2f a, b;
        a.x = arow[k];     a.y = arow[k + 1];
        b.x = wrow[k];     b.y = wrow[k + 1];
        // 8 args: (neg_a, A, neg_b, B, c_mod, C, reuse_a, reuse_b)
        c = __builtin_amdgcn_wmma_f32_16x16x4_f32(false, a, false, b,
                                                  (short)0, c, false, false);
    }

    int rowbase = tm * 16 + 8 * half;
#pragma unroll
    for (int v = 0; v < 8; ++v) {
        float r = c[v];
        if (do_relu) r = fmaxf(r, 0.0f);
        out[(size_t)(rowbase + v) * dout + ocol] = r;
    }
}

// ---------------- driver ----------------

extern "C" void kernel_launch(void* const* d_in, const int* in_sizes, int n_in,
                              void* d_out, int out_size, void* d_ws, size_t ws_size,
                              hipStream_t stream) {
    (void)in_sizes; (void)n_in; (void)out_size; (void)ws_size;

    const int*   row = (const int*)d_in[0];          // edge_index[0], int32
    const int*   col = row + NEDGES;                 // edge_index[1]
    // d_in[1] = edge_weight (unused by reference message())
    const float* x   = (const float*)d_in[2];
    const unsigned char* M = (const unsigned char*)d_in[3];   // numpy bool, 1 byte
    const float* Wl[3] = {(const float*)d_in[4],  (const float*)d_in[7],  (const float*)d_in[10]};
    const float* bl[3] = {(const float*)d_in[5],  (const float*)d_in[8],  (const float*)d_in[11]};
    const float* zl[3] = {(const float*)d_in[6],  (const float*)d_in[9],  (const float*)d_in[12]};
    const float* Wf = (const float*)d_in[13];
    const float* bf = (const float*)d_in[14];

    float* ws   = (float*)d_ws;
    float* dinv = ws;                         // N          (deg -> d^-1/2 in place)
    float* norm = dinv + NNODES;              // E
    float* xt   = norm + NEDGES;              // N*D
    float* agg  = xt   + (size_t)NNODES*DIM;  // N*D
    float* h    = agg  + (size_t)NNODES*DIM;  // N*D

    const int T  = 256;
    const int ND = NNODES * DIM;

    // degrees -> d^{-1/2} -> per-edge norm
    zero_f32  <<<(NNODES + T - 1) / T, T, 0, stream>>>(dinv, NNODES);
    deg_kernel<<<(NEDGES + T - 1) / T, T, 0, stream>>>(col, dinv, NEDGES);
    dinv_kernel<<<(NNODES + T - 1) / T, T, 0, stream>>>(dinv, NNODES);
    norm_kernel<<<(NEDGES + T - 1) / T, T, 0, stream>>>(row, col, dinv, norm, NEDGES);

    const int tiles_m = NNODES / 16;          // 3125 (exact)
    for (int l = 0; l < 3; ++l) {
        xtilde_kernel<<<(ND + T - 1) / T, T, 0, stream>>>(M, x, h, xt, ND, l > 0 ? 1 : 0);
        zero_f32     <<<(ND + T - 1) / T, T, 0, stream>>>(agg, ND);
        scatter_kernel<<<(NEDGES * 32 + T - 1) / T, T, 0, stream>>>(row, col, norm, xt, agg, NEDGES);
        int tiles_o = DIM / 16;               // 8
        int waves   = tiles_m * tiles_o;
        gemm_wmma_f32<<<(waves * 32 + T - 1) / T, T, 0, stream>>>(
            agg, Wl[l], bl[l], zl[l], h, tiles_m, tiles_o, DIM, DIM, 1);
    }

    // final: out = h @ Wf^T + bf  (no relu)
    {
        int tiles_o = DOUTF / 16;             // 4
        int waves   = tiles_m * tiles_o;
        gemm_wmma_f32<<<(waves * 32 + T - 1) / T, T, 0, stream>>>(
            h, Wf, bf, nullptr, (float*)d_out, tiles_m, tiles_o, DIM, DOUTF, 0);
    }
}